// NdLinear_88356067213436
// MI455X (gfx1250) — compile-verified
//
#include <hip/hip_runtime.h>

// Low-rank linear: out = x @ (U @ V) + bias, factored as
//   Stage 1: T = x @ U          [8192 x 64]   (split-K x4 partials in d_ws, needs 8 MB)
//   Stage 2: out = (sum T_p) @ V + bias  [8192 x 4096]
// fp32 WMMA (v_wmma_f32_16x16x4_f32) preserves reference fp32 precision.
//
// B-operand staging uses a K-pair-interleaved LDS layout:
//   lds[(k/2)*LSTR + 2*n + (k&1)]
// so each WMMA B operand (rows k, k+1 at column n) is one contiguous
// ds_load_b64 into an even-aligned VGPR pair (no repacking movs).
// LSTR = 160 words (160 % 64 == 32): the two half-waves (kpair, kpair+1)
// read complementary 32-bank sets -> conflict-free b64 LDS reads.

typedef float v2f __attribute__((ext_vector_type(2)));
typedef float v4f __attribute__((ext_vector_type(4)));
typedef float v8f __attribute__((ext_vector_type(8)));

constexpr int IN_F   = 4096;
constexpr int OUT_F  = 4096;
constexpr int RANK   = 64;
constexpr int NTOK   = 8192;
constexpr int KSPLIT = 4;        // stage-1 split-K factor (occupancy)
constexpr int KCHUNK = 128;      // K rows of U staged in LDS per iteration
constexpr int LSTR   = 160;      // interleaved LDS row stride per K-pair (words)

// ---------------------------------------------------------------------------
// Stage 1: Tpart[p] = x[:, p*1024:(p+1)*1024] @ U[p*1024:(p+1)*1024, :]
// Block = 256 threads = 8 waves; each wave: one 16-row M-tile x all 64 N cols.
// ---------------------------------------------------------------------------
__global__ __launch_bounds__(256) void lowrank_xu(const float* __restrict__ x,
                                                  const float* __restrict__ U,
                                                  float* __restrict__ Tpart) {
  __shared__ float lU[(KCHUNK / 2) * LSTR];         // 40 KB of 320 KB WGP LDS

  const int tid  = threadIdx.x;
  const int lane = tid & 31;
  const int ln16 = lane & 15;                       // M (A/C) or N (B) within tile
  const int hi   = lane >> 4;                       // half-wave selector
  const int wave = tid >> 5;
  const int mbase = blockIdx.x * 128 + wave * 16;
  const int kpart = blockIdx.y * (IN_F / KSPLIT);

  const float* __restrict__ xrow = x + (size_t)(mbase + ln16) * IN_F + kpart;

  v8f acc[4] = {};                                  // 16x64 fp32 accumulator tile

  for (int kb = 0; kb < IN_F / KSPLIT; kb += KCHUNK) {
    __syncthreads();
    // Cooperatively stage U[kpart+kb .. +128, 0:64] into interleaved LDS.
    // Global side stays b128-coalesced; LDS side scatters 4 dwords.
    for (int i = tid; i < KCHUNK * (RANK / 4); i += 256) {
      const int r  = i >> 4;                        // U row within chunk
      const int c4 = i & 15;                        // float4 column
      const v4f u = *(const v4f*)(U + (size_t)(kpart + kb + r) * RANK + c4 * 4);
      float* dst = &lU[(r >> 1) * LSTR + (r & 1)];
#pragma unroll
      for (int j = 0; j < 4; ++j) dst[2 * (c4 * 4 + j)] = u[j];
    }
    __syncthreads();

#pragma unroll 4
    for (int k0 = 0; k0 < KCHUNK; k0 += 4) {
      // A (16x4 f32): lane ln16 = row M, halves hold K pairs {0,1}/{2,3}; 8B coalesced
      const v2f a = *(const v2f*)(xrow + kb + k0 + 2 * hi);
#pragma unroll
      for (int t = 0; t < 4; ++t) {
        // B (4x16 f32): rows (k0+2hi, k0+2hi+1) at col n -> one ds_load_b64
        const v2f b = *(const v2f*)(
            &lU[((k0 >> 1) + hi) * LSTR + 2 * (t * 16 + ln16)]);
        acc[t] = __builtin_amdgcn_wmma_f32_16x16x4_f32(
            false, a, false, b, (short)0, acc[t], false, false);
      }
    }
  }

  // C layout: VGPR r -> M = r + 8*hi, lane ln16 -> N
  float* __restrict__ Tout = Tpart + (size_t)blockIdx.y * NTOK * RANK;
#pragma unroll
  for (int t = 0; t < 4; ++t)
#pragma unroll
    for (int r = 0; r < 8; ++r)
      Tout[(size_t)(mbase + hi * 8 + r) * RANK + t * 16 + ln16] = acc[t][r];
}

// ---------------------------------------------------------------------------
// Stage 2: out = (sum_p Tpart[p]) @ V + bias
// Block = 8 waves: 8 M-tiles (128 rows) x one 64-wide N strip; V strip in LDS.
// ---------------------------------------------------------------------------
__global__ __launch_bounds__(256) void lowrank_tv(const float* __restrict__ Tpart,
                                                  const float* __restrict__ V,
                                                  const float* __restrict__ bias,
                                                  float* __restrict__ out) {
  __shared__ float lV[(RANK / 2) * LSTR];           // 20 KB

  const int tid  = threadIdx.x;
  const int lane = tid & 31;
  const int ln16 = lane & 15;
  const int hi   = lane >> 4;
  const int wave = tid >> 5;
  const int mbase = blockIdx.x * 128 + wave * 16;
  const int nbase = blockIdx.y * 64;

  // Stage V[0:64, nbase:nbase+64] into interleaved LDS (64 rows * 16 float4).
  for (int i = tid; i < RANK * (64 / 4); i += 256) {
    const int r  = i >> 4;
    const int c4 = i & 15;
    const v4f v = *(const v4f*)(V + (size_t)r * OUT_F + nbase + c4 * 4);
    float* dst = &lV[(r >> 1) * LSTR + (r & 1)];
#pragma unroll
    for (int j = 0; j < 4; ++j) dst[2 * (c4 * 4 + j)] = v[j];
  }
  __syncthreads();

  const size_t trow = (size_t)(mbase + ln16) * RANK;

  v8f acc[4] = {};
#pragma unroll
  for (int k0 = 0; k0 < RANK; k0 += 4) {
    const int koff = k0 + 2 * hi;
    // A operand = sum of split-K partials (tiny, L2-resident reads)
    v2f a = *(const v2f*)(Tpart + trow + koff);
#pragma unroll
    for (int p = 1; p < KSPLIT; ++p)
      a += *(const v2f*)(Tpart + (size_t)p * NTOK * RANK + trow + koff);
#pragma unroll
    for (int t = 0; t < 4; ++t) {
      const v2f b = *(const v2f*)(
          &lV[((k0 >> 1) + hi) * LSTR + 2 * (t * 16 + ln16)]);
      acc[t] = __builtin_amdgcn_wmma_f32_16x16x4_f32(
          false, a, false, b, (short)0, acc[t], false, false);
    }
  }

#pragma unroll
  for (int t = 0; t < 4; ++t) {
    const float bv = bias[nbase + t * 16 + ln16];
#pragma unroll
    for (int r = 0; r < 8; ++r)
      out[(size_t)(mbase + hi * 8 + r) * OUT_F + nbase + t * 16 + ln16] =
          acc[t][r] + bv;
  }
}

extern "C" void kernel_launch(void* const* d_in, const int* in_sizes, int n_in,
                              void* d_out, int out_size, void* d_ws, size_t ws_size,
                              hipStream_t stream) {
  const float* x    = (const float*)d_in[0];   // [8192, 4096]
  const float* U    = (const float*)d_in[1];   // [4096, 64]
  const float* V    = (const float*)d_in[2];   // [64, 4096]
  const float* bias = (const float*)d_in[3];   // [4096]
  float* out   = (float*)d_out;                // [8192, 4096]
  float* Tpart = (float*)d_ws;                 // KSPLIT x [8192 x 64] fp32 = 8 MB

  lowrank_xu<<<dim3(NTOK / 128, KSPLIT), 256, 0, stream>>>(x, U, Tpart);
  lowrank_tv<<<dim3(NTOK / 128, OUT_F / 64), 256, 0, stream>>>(Tpart, V, bias, out);
}